// GCN_link_54348516164018
// MI455X (gfx1250) — compile-verified
//
#include <hip/hip_runtime.h>
#include <hip/hip_bf16.h>

typedef __attribute__((ext_vector_type(2))) float v2f;
typedef __attribute__((ext_vector_type(8))) float v8f;

// ---------------------------------------------------------------------------
// CSR construction: per-node in-degree counts, scan, fill
// ---------------------------------------------------------------------------
__global__ void cnt_zero_k(int* cnt, int N) {
    int i = blockIdx.x * blockDim.x + threadIdx.x;
    if (i < N) cnt[i] = 0;
}

__global__ void cnt_count_k(const int* __restrict__ ei, int* cnt, int E) {
    int e = blockIdx.x * blockDim.x + threadIdx.x;
    if (e < E) atomicAdd(&cnt[ei[E + e]], 1);  // col endpoint receives
}

// dis[n] = rsqrt(deg[n]), deg = cnt + 1 (self-loop)
__global__ void dis_k(const int* __restrict__ cnt, float* dis, int N) {
    int i = blockIdx.x * blockDim.x + threadIdx.x;
    if (i < N) dis[i] = rsqrtf((float)(cnt[i] + 1));
}

// Single-block exclusive scan of cnt[0..N) -> rowptr[0..N], cursor copy.
// N=50000: 49 chunks of 1024, Hillis-Steele in shared memory. Few us, fine.
__global__ void scan_k(const int* __restrict__ cnt, int* rowptr, int* cursor, int N) {
    __shared__ int sdata[1024];
    __shared__ int s_running;
    const int tid = threadIdx.x;
    if (tid == 0) s_running = 0;
    __syncthreads();
    for (int base = 0; base < N; base += 1024) {
        int i = base + tid;
        int v = (i < N) ? cnt[i] : 0;
        sdata[tid] = v;
        __syncthreads();
        for (int off = 1; off < 1024; off <<= 1) {
            int t = (tid >= off) ? sdata[tid - off] : 0;
            __syncthreads();
            sdata[tid] += t;
            __syncthreads();
        }
        int excl = sdata[tid] - v;
        int p = s_running + excl;
        if (i < N) { rowptr[i] = p; cursor[i] = p; }
        __syncthreads();                       // all reads of s_running done
        if (tid == 1023) s_running += sdata[1023];
        __syncthreads();
    }
    if (tid == 0) rowptr[N] = s_running;       // == E
}

// srcrow[slot] = row, slot claimed with one int atomic per edge.
__global__ void fill_k(const int* __restrict__ ei, int* cursor,
                       int* __restrict__ srcrow, int E) {
    int e = blockIdx.x * blockDim.x + threadIdx.x;
    if (e < E) {
        int r = ei[e];
        int c = ei[E + e];
        int p = atomicAdd(&cursor[c], 1);
        srcrow[p] = r;
    }
}

// ---------------------------------------------------------------------------
// Dense GEMM via V_WMMA_F32_16X16X4_F32 (wave32, 16x16 C tile per wave).
// A: [N x K] row-major, W: [K x M] row-major, C: [N x M] row-major.
// Requires N % 16 == 0 (50000 = 3125 * 16), K % 4 == 0.
// ---------------------------------------------------------------------------
template <int K, int M>
__global__ void gemm_wmma_k(const float* __restrict__ A,
                            const float* __restrict__ W,
                            float* __restrict__ C, int N) {
    const int lane = threadIdx.x & 31;
    const int wave = threadIdx.x >> 5;
    const int half = lane >> 4;
    const int l16  = lane & 15;
    const int m0 = blockIdx.x * 16;
    const int n0 = wave * 16;
    if (m0 >= N) return;

    const float* arow = A + (size_t)(m0 + l16) * K;

    v8f acc = {};
#pragma unroll 8
    for (int k0 = 0; k0 < K; k0 += 4) {
        const int ka = k0 + 2 * half;
        v2f a, b;
        a.x = arow[ka];
        a.y = arow[ka + 1];
        b.x = W[(size_t)ka * M + n0 + l16];
        b.y = W[(size_t)(ka + 1) * M + n0 + l16];
        acc = __builtin_amdgcn_wmma_f32_16x16x4_f32(
            false, a, false, b, (short)0, acc, false, false);
    }

#pragma unroll
    for (int r = 0; r < 8; ++r) {
        C[(size_t)(m0 + r + 8 * half) * M + n0 + l16] = acc[r];
    }
}

// ---------------------------------------------------------------------------
// Gather-reduce aggregation (replaces scatter atomics):
// one wave per destination node; lane holds M/32 features in registers.
//   out[n] = (opt relu)( h[n]*dis[n]^2 + bias + sum_{r in adj(n)} h[r]*dis[r]*dis[n] )
// Edge rows batched 32/wave and broadcast via __shfl so srcrow/dis loads are
// coalesced; h rows are 512B/256B L2-resident bursts.
// ---------------------------------------------------------------------------
template <int M, bool RELU>
__global__ void gather_agg_k(const float* __restrict__ hw,
                             const float* __restrict__ dis,
                             const float* __restrict__ bias,
                             const int* __restrict__ rowptr,
                             const int* __restrict__ srcrow,
                             float* __restrict__ out, int N) {
    constexpr int VPT = M / 32;  // floats per lane (4 for M=128, 2 for M=64)
    const int lane = threadIdx.x & 31;
    const int n = blockIdx.x * (blockDim.x >> 5) + (threadIdx.x >> 5);
    if (n >= N) return;

    const float dn = dis[n];
    float acc[VPT];
    {
        const float* hrow = hw + (size_t)n * M + lane * VPT;
#pragma unroll
        for (int v = 0; v < VPT; ++v)
            acc[v] = hrow[v] * dn * dn + bias[lane * VPT + v];
    }

    const int beg = rowptr[n];
    const int end = rowptr[n + 1];
    for (int j0 = beg; j0 < end; j0 += 32) {
        const int nj = min(32, end - j0);
        int   rj = 0;
        float dj = 0.0f;
        if (j0 + lane < end) {
            rj = srcrow[j0 + lane];
            dj = dis[rj];
        }
        for (int i = 0; i < nj; ++i) {
            const int   r   = __shfl(rj, i);
            const float nrm = __shfl(dj, i) * dn;
            const float* hr = hw + (size_t)r * M + lane * VPT;
#pragma unroll
            for (int v = 0; v < VPT; ++v)
                acc[v] = fmaf(hr[v], nrm, acc[v]);
        }
    }

    float* o = out + (size_t)n * M + lane * VPT;
#pragma unroll
    for (int v = 0; v < VPT; ++v)
        o[v] = RELU ? fmaxf(acc[v], 0.0f) : acc[v];
}

// ---------------------------------------------------------------------------
// Launch
// ---------------------------------------------------------------------------
extern "C" void kernel_launch(void* const* d_in, const int* in_sizes, int n_in,
                              void* d_out, int out_size, void* d_ws, size_t ws_size,
                              hipStream_t stream) {
    constexpr int DIN = 128, DHID = 128, DOUT = 64;

    const float* x  = (const float*)d_in[0];
    const int*   ei = (const int*)d_in[1];
    // d_in[2] edge_label, d_in[3] edge_label_index: unused by reference output
    const float* W1 = (const float*)d_in[4];
    const float* b1 = (const float*)d_in[5];
    const float* W2 = (const float*)d_in[6];
    const float* b2 = (const float*)d_in[7];

    const int N = in_sizes[0] / DIN;   // 50000
    const int E = in_sizes[1] / 2;     // 800000

    // Workspace layout (256B aligned slices)
    auto align256 = [](size_t b) { return (b + 255) & ~(size_t)255; };
    char* w = (char*)d_ws;
    size_t off = 0;
    int*   cnt    = (int*)(w + off); off = align256(off + (size_t)N * 4);
    int*   rowptr = (int*)(w + off); off = align256(off + (size_t)(N + 1) * 4);
    int*   cursor = (int*)(w + off); off = align256(off + (size_t)N * 4);
    int*   srcrow = (int*)(w + off); off = align256(off + (size_t)E * 4);
    float* dis    = (float*)(w + off); off = align256(off + (size_t)N * 4);
    float* hbuf   = (float*)(w + off); off = align256(off + (size_t)N * DHID * 4);
    float* agg1   = (float*)(w + off); off = align256(off + (size_t)N * DHID * 4);
    float* z      = (float*)d_out;

    const int T = 256;
    const int nodeB = (N + T - 1) / T;
    const int edgeB = (E + T - 1) / T;

    // 1) CSR build + normalization
    cnt_zero_k<<<nodeB, T, 0, stream>>>(cnt, N);
    cnt_count_k<<<edgeB, T, 0, stream>>>(ei, cnt, E);
    dis_k<<<nodeB, T, 0, stream>>>(cnt, dis, N);
    scan_k<<<1, 1024, 0, stream>>>(cnt, rowptr, cursor, N);
    fill_k<<<edgeB, T, 0, stream>>>(ei, cursor, srcrow, E);

    // 2) layer 1: h1 = x @ W1 (WMMA)
    gemm_wmma_k<DIN, DHID><<<N / 16, (DHID / 16) * 32, 0, stream>>>(x, W1, hbuf, N);

    // 3) layer 1 aggregation: gather-reduce + bias + ReLU, no f32 atomics
    {
        const int wavesPerBlk = T / 32;
        gather_agg_k<DHID, true><<<(N + wavesPerBlk - 1) / wavesPerBlk, T, 0, stream>>>(
            hbuf, dis, b1, rowptr, srcrow, agg1, N);
    }

    // 4) layer 2: h2 = relu_out @ W2 (WMMA), reuse hbuf
    gemm_wmma_k<DHID, DOUT><<<N / 16, (DOUT / 16) * 32, 0, stream>>>(agg1, W2, hbuf, N);

    // 5) layer 2 aggregation straight into d_out (bias folded, no ReLU)
    {
        const int wavesPerBlk = T / 32;
        gather_agg_k<DOUT, false><<<(N + wavesPerBlk - 1) / wavesPerBlk, T, 0, stream>>>(
            hbuf, dis, b2, rowptr, srcrow, z, N);
    }
}